// Decoder_gru_2_8589935086
// MI455X (gfx1250) — compile-verified
//
#include <hip/hip_runtime.h>
#include <math.h>

// ---------------------------------------------------------------------------
// Decoder_gru_2 for MI455X (gfx1250, wave32).
//
// Key algorithmic move: gi/gh of the GRUCell depend only on x-rows, so we
// precompute Gi = x@W_ih^T + b_ih and Gh = x@W_hh^T + b_hh as [84,192] once
// (4 MFLOP instead of 171 MFLOP of per-pair GEMM). Full-tensor LayerNorms
// force grid-wide reductions -> kernel boundaries; reductions are wave-local
// shuffles + 2 float atomicAdds per wave.
//
// All GEMMs use V_WMMA_F32_16X16X4_F32 (fp32 in / fp32 acc) so the fp32
// reference tolerance is respected. One wave per 16x16 output tile.
// ---------------------------------------------------------------------------

typedef float v2f __attribute__((ext_vector_type(2)));
typedef float v8f __attribute__((ext_vector_type(8)));

static constexpr int Nn = 84;                 // nodes
static constexpr int Hh = 64;                 // hidden
static constexpr int Mp = Nn * (Nn - 1) / 2;  // 3486 pairs
static constexpr float LNEPS = 1e-5f;

// ---------------- kernel 0: init (zero out, zero reductions, triu indices) --
__global__ void k_init(int* __restrict__ ii, int* __restrict__ jj,
                       float* __restrict__ red, float* __restrict__ out) {
  int t = blockIdx.x * blockDim.x + threadIdx.x;
  if (t < 8) red[t] = 0.0f;
  if (t < Nn * Nn) out[t] = 0.0f;
  if (t < Mp) {
    int mm = t, i = 0;
    while (mm >= Nn - 1 - i) { mm -= Nn - 1 - i; ++i; }
    ii[t] = i;
    jj[t] = i + 1 + mm;
  }
}

// ---------------- kernel 1: Gi/Gh = x @ {W_ih,W_hh}^T + bias  (WMMA f32) ----
// 6 row tiles (96>=84 rows) x 24 col tiles (12 Gi + 12 Gh), one wave each.
__global__ void k_gates(const float* __restrict__ x,
                        const float* __restrict__ Wih, const float* __restrict__ Whh,
                        const float* __restrict__ bih, const float* __restrict__ bhh,
                        float* __restrict__ Gi, float* __restrict__ Gh) {
  int wave = (blockIdx.x * blockDim.x + threadIdx.x) >> 5;
  int lane = threadIdx.x & 31;
  const int rowTiles = 6, colTiles = 24;
  if (wave >= rowTiles * colTiles) return;   // wave-uniform

  int rt = wave / colTiles, ct = wave % colTiles;
  bool isGh = (ct >= 12);
  int ctl = isGh ? ct - 12 : ct;
  const float* W = isGh ? Whh : Wih;
  const float* b = isGh ? bhh : bih;
  float* G = isGh ? Gh : Gi;

  int row0 = rt * 16, col0 = ctl * 16;
  int r = lane & 15, kh = lane >> 4;
  int arow = min(row0 + r, Nn - 1);          // clamp pad rows (masked on store)

  v8f c = {};
  for (int k0 = 0; k0 < 64; k0 += 4) {
    int ka = k0 + 2 * kh;
    v2f a, bb;
    a.x = x[arow * 64 + ka];
    a.y = x[arow * 64 + ka + 1];
    bb.x = W[(col0 + r) * 64 + ka];          // B(k,n) = W[n*64+k]  (Y = X W^T)
    bb.y = W[(col0 + r) * 64 + ka + 1];
    c = __builtin_amdgcn_wmma_f32_16x16x4_f32(false, a, false, bb, (short)0, c,
                                              false, false);
  }
  int n = col0 + r;
  float bias = b[n];
  int mbase = row0 + kh * 8;
  for (int v = 0; v < 8; ++v) {
    int m = mbase + v;
    if (m < Nn) G[m * 192 + n] = c[v] + bias;
  }
}

// ---------------- kernel 2: per-pair GRU gate math (elementwise) ------------
__global__ void k_gru(const float* __restrict__ x,
                      const float* __restrict__ Gi, const float* __restrict__ Gh,
                      const int* __restrict__ ii, const int* __restrict__ jj,
                      float* __restrict__ h) {
  int t = blockIdx.x * blockDim.x + threadIdx.x;
  if (t >= Mp * 64) return;
  int m = t >> 6, c = t & 63;
  int i = ii[m], j = jj[m];
  float gr = Gi[i * 192 + c]       + Gh[j * 192 + c];
  float gz = Gi[i * 192 + 64 + c]  + Gh[j * 192 + 64 + c];
  float gin = Gi[i * 192 + 128 + c];
  float ghn = Gh[j * 192 + 128 + c];
  float rg = 1.0f / (1.0f + expf(-gr));
  float zg = 1.0f / (1.0f + expf(-gz));
  float ng = tanhf(gin + rg * ghn);
  float xv = x[j * 64 + c];
  h[t] = (1.0f - zg) * ng + zg * xv;
}

// ---------------- fused (optional LN-apply) -> Linear -> ReLU -> reduce -----
// Y[M,NOUT] = relu( norm(Xin[M,K]) @ W[NOUT,K]^T + bias ),
// norm(v) = (v-mean)*rstd*g + be using scalars derived from redIn[0..1].
// Accumulates sum/sumsq of valid outputs into redOut[0], redOut[1].
template <int K, int NOUT, bool DO_NORM>
__global__ void k_linear(const float* __restrict__ Xin,
                         const float* __restrict__ W, const float* __restrict__ bias,
                         const float* __restrict__ g, const float* __restrict__ be,
                         const float* __restrict__ redIn, float invCnt,
                         float* __restrict__ Y, float* __restrict__ redOut) {
  const int rowTiles = (Mp + 15) / 16;   // 218
  const int colTiles = NOUT / 16;
  int wave = (blockIdx.x * blockDim.x + threadIdx.x) >> 5;
  int lane = threadIdx.x & 31;

  float s = 0.0f, s2 = 0.0f;
  bool active = (wave < rowTiles * colTiles);  // wave-uniform
  if (active) {
    int rt = wave / colTiles, ct = wave % colTiles;
    int row0 = rt * 16, col0 = ct * 16;
    int r = lane & 15, kh = lane >> 4;
    int arow = min(row0 + r, Mp - 1);

    float mean = 0.0f, rstd = 1.0f;
    if (DO_NORM) {
      float su = redIn[0], sq = redIn[1];
      mean = su * invCnt;
      float var = sq * invCnt - mean * mean;
      rstd = rsqrtf(var + LNEPS);
    }

    v8f c = {};
    for (int k0 = 0; k0 < K; k0 += 4) {
      int ka = k0 + 2 * kh;
      float a0 = Xin[arow * K + ka];
      float a1 = Xin[arow * K + ka + 1];
      if (DO_NORM) {
        a0 = (a0 - mean) * rstd * g[arow * K + ka]     + be[arow * K + ka];
        a1 = (a1 - mean) * rstd * g[arow * K + ka + 1] + be[arow * K + ka + 1];
      }
      v2f a, bb;
      a.x = a0; a.y = a1;
      bb.x = W[(col0 + r) * K + ka];
      bb.y = W[(col0 + r) * K + ka + 1];
      c = __builtin_amdgcn_wmma_f32_16x16x4_f32(false, a, false, bb, (short)0, c,
                                                false, false);
    }
    int n = col0 + r;
    float bv = bias[n];
    int mbase = row0 + kh * 8;
    for (int v = 0; v < 8; ++v) {
      int m = mbase + v;
      if (m < Mp) {
        float y = c[v] + bv;
        y = y > 0.0f ? y : 0.0f;             // ReLU
        Y[m * NOUT + n] = y;
        s += y;
        s2 += y * y;
      }
    }
  }
  // wave32 butterfly reduction, then 2 atomics per wave
  for (int off = 16; off > 0; off >>= 1) {
    s  += __shfl_xor(s, off, 32);
    s2 += __shfl_xor(s2, off, 32);
  }
  if (active && lane == 0) {
    atomicAdd(&redOut[0], s);
    atomicAdd(&redOut[1], s2);
  }
}

// ---------------- kernel 6: LN3 -> Linear4 -> sigmoid -> symmetric scatter --
__global__ void k_final(const float* __restrict__ y3,
                        const float* __restrict__ g3, const float* __restrict__ be3,
                        const float* __restrict__ W4, const float* __restrict__ b4,
                        const float* __restrict__ red,
                        const int* __restrict__ ii, const int* __restrict__ jj,
                        float* __restrict__ out) {
  int m = blockIdx.x * blockDim.x + threadIdx.x;
  if (m >= Mp) return;
  float invCnt = 1.0f / (float)(Mp * 32);
  float mean = red[4] * invCnt;
  float var = red[5] * invCnt - mean * mean;
  float rstd = rsqrtf(var + LNEPS);
  float acc = b4[0];
  for (int c = 0; c < 32; ++c) {
    float v = (y3[m * 32 + c] - mean) * rstd * g3[m * 32 + c] + be3[m * 32 + c];
    acc += v * W4[c];
  }
  float o = 1.0f / (1.0f + expf(-acc));
  int i = ii[m], j = jj[m];
  out[i * Nn + j] = o;
  out[j * Nn + i] = o;
}

// ---------------------------------------------------------------------------
extern "C" void kernel_launch(void* const* d_in, const int* in_sizes, int n_in,
                              void* d_out, int out_size, void* d_ws, size_t ws_size,
                              hipStream_t stream) {
  const float* x    = (const float*)d_in[0];
  const float* Wih  = (const float*)d_in[1];
  const float* Whh  = (const float*)d_in[2];
  const float* bih  = (const float*)d_in[3];
  const float* bhh  = (const float*)d_in[4];
  const float* W1   = (const float*)d_in[5];
  const float* b1   = (const float*)d_in[6];
  const float* g1   = (const float*)d_in[7];
  const float* be1  = (const float*)d_in[8];
  const float* W2   = (const float*)d_in[9];
  const float* b2   = (const float*)d_in[10];
  const float* g2   = (const float*)d_in[11];
  const float* be2  = (const float*)d_in[12];
  const float* W3   = (const float*)d_in[13];
  const float* b3   = (const float*)d_in[14];
  const float* g3   = (const float*)d_in[15];
  const float* be3  = (const float*)d_in[16];
  const float* W4   = (const float*)d_in[17];
  const float* b4   = (const float*)d_in[18];
  float* out = (float*)d_out;

  // workspace layout (fp32 slots; idx arrays reuse as int)
  float* ws = (float*)d_ws;
  int*   ii  = (int*)ws;
  int*   jj  = ii + Mp;
  float* Gi  = ws + 2 * Mp;
  float* Gh  = Gi + Nn * 192;
  float* h   = Gh + Nn * 192;
  float* y1  = h  + Mp * 64;
  float* y2  = y1 + Mp * 64;
  float* y3  = y2 + Mp * 32;
  float* red = y3 + Mp * 32;       // red[0..5]: (sum,sumsq) x 3 LN stages

  // 0) init: zero d_out (diagonal must be 0), zero reductions, triu indices
  k_init<<<(Nn * Nn + 255) / 256, 256, 0, stream>>>(ii, jj, red, out);

  // 1) Gi/Gh gate pre-GEMM: 6x24 tiles = 144 waves = 18 blocks of 8 waves
  k_gates<<<18, 256, 0, stream>>>(x, Wih, Whh, bih, bhh, Gi, Gh);

  // 2) per-pair GRU gates (elementwise over M*64)
  k_gru<<<(Mp * 64 + 255) / 256, 256, 0, stream>>>(x, Gi, Gh, ii, jj, h);

  // 3) y1 = relu(h @ W1^T + b1), reduce -> red[0..1]   (218*4 tiles)
  k_linear<64, 64, false><<<(218 * 4) / 8, 256, 0, stream>>>(
      h, W1, b1, nullptr, nullptr, nullptr, 0.0f, y1, red + 0);

  // 4) y2 = relu(LN1(y1) @ W2^T + b2), reduce -> red[2..3]   (218*2 tiles)
  k_linear<64, 32, true><<<(218 * 2 + 7) / 8, 256, 0, stream>>>(
      y1, W2, b2, g1, be1, red + 0, 1.0f / (float)(Mp * 64), y2, red + 2);

  // 5) y3 = relu(LN2(y2) @ W3^T + b3), reduce -> red[4..5]
  k_linear<32, 32, true><<<(218 * 2 + 7) / 8, 256, 0, stream>>>(
      y2, W3, b3, g2, be2, red + 2, 1.0f / (float)(Mp * 32), y3, red + 4);

  // 6) o = sigmoid(LN3(y3) @ W4^T + b4); scatter A[i,j]=A[j,i]=o
  k_final<<<(Mp + 255) / 256, 256, 0, stream>>>(
      y3, g3, be3, W4, b4, red, ii, jj, out);
}